// QMessagePassing_44272522887312
// MI455X (gfx1250) — compile-verified
//
#include <hip/hip_runtime.h>
#include <math.h>

typedef __attribute__((ext_vector_type(2))) float v2f;
typedef __attribute__((ext_vector_type(8))) float v8f;

// ---- workspace layout (float offsets) ----
#define WS_EDGE_UR 0        // 256 f
#define WS_EDGE_UI 256      // 256 f
#define WS_UPD_UR  512      // 256 f
#define WS_UPD_UI  768      // 256 f
#define WS_AMP     1024     // 38 float2 = 76 f (rounded up)
#define WS_PART    1152     // 256 f
#define WS_PSI     1408     // 2^20 f = float2[2^19], 8B aligned

__device__ inline float2 cmul(float2 a, float2 b) {
    return make_float2(fmaf(a.x, b.x, -a.y * b.y), fmaf(a.x, b.y, a.y * b.x));
}
__device__ inline float2 cadd(float2 a, float2 b) { return make_float2(a.x + b.x, a.y + b.y); }

// ================= gate-matrix construction (1 block, 128 threads) =================
__device__ void mat_identity(float2* U, int tid) {
    for (int e = tid; e < 256; e += 128) {
        U[e].x = ((e >> 4) == (e & 15)) ? 1.f : 0.f;
        U[e].y = 0.f;
    }
    __syncthreads();
}

// left-multiply U by 1q gate embedded at local bit p (16 cols x 8 row pairs = 128 threads)
__device__ void apply1q(float2* U, int tid, int p,
                        float2 g00, float2 g01, float2 g10, float2 g11) {
    int c  = tid & 15;
    int pr = tid >> 4; // 0..7
    int low = pr & ((1 << p) - 1);
    int r0  = ((pr >> p) << (p + 1)) | low;
    int r1  = r0 | (1 << p);
    float2 x0 = U[r0 * 16 + c], x1 = U[r1 * 16 + c];
    U[r0 * 16 + c] = cadd(cmul(g00, x0), cmul(g01, x1));
    U[r1 * 16 + c] = cadd(cmul(g10, x0), cmul(g11, x1));
    __syncthreads();
}

// controlled 1q gate: control bit pc, target bit pt (16 cols x 4 pairs = 64 threads)
__device__ void applyCtrl(float2* U, int tid, int pc, int pt,
                          float2 g00, float2 g01, float2 g10, float2 g11) {
    if (tid < 64) {
        int c  = tid & 15;
        int pr = tid >> 4; // 0..3
        int fb[2], nf = 0;
        for (int b = 0; b < 4; b++)
            if (b != pc && b != pt) fb[nf++] = b;
        int r0 = ((pr & 1) << fb[0]) | (((pr >> 1) & 1) << fb[1]) | (1 << pc);
        int r1 = r0 | (1 << pt);
        float2 x0 = U[r0 * 16 + c], x1 = U[r1 * 16 + c];
        U[r0 * 16 + c] = cadd(cmul(g00, x0), cmul(g01, x1));
        U[r1 * 16 + c] = cadd(cmul(g10, x0), cmul(g11, x1));
    }
    __syncthreads();
}

__device__ void rx_gate(float t, float2& g00, float2& g01, float2& g10, float2& g11) {
    float c = cosf(t * 0.5f), s = sinf(t * 0.5f);
    g00 = make_float2(c, 0.f); g01 = make_float2(0.f, -s);
    g10 = make_float2(0.f, -s); g11 = make_float2(c, 0.f);
}
__device__ void ry_gate(float t, float2& g00, float2& g01, float2& g10, float2& g11) {
    float c = cosf(t * 0.5f), s = sinf(t * 0.5f);
    g00 = make_float2(c, 0.f); g01 = make_float2(-s, 0.f);
    g10 = make_float2(s, 0.f); g11 = make_float2(c, 0.f);
}
__device__ void rz_gate(float t, float2& g00, float2& g01, float2& g10, float2& g11) {
    float c = cosf(t * 0.5f), s = sinf(t * 0.5f);
    g00 = make_float2(c, -s); g01 = make_float2(0.f, 0.f);
    g10 = make_float2(0.f, 0.f); g11 = make_float2(c, s);
}

// StronglyEntanglingLayers depth=1 on local bits (w0,w1,w2); p has 9 angles
__device__ void strong3(float2* U, int tid, int w0, int w1, int w2, const float* p) {
    const int w[3] = {w0, w1, w2};
    for (int qi = 0; qi < 3; qi++) {
        float2 a, b, c2, d;
        rz_gate(p[qi * 3 + 0], a, b, c2, d); apply1q(U, tid, w[qi], a, b, c2, d);
        ry_gate(p[qi * 3 + 1], a, b, c2, d); apply1q(U, tid, w[qi], a, b, c2, d);
        rx_gate(p[qi * 3 + 2], a, b, c2, d); apply1q(U, tid, w[qi], a, b, c2, d);
    }
    float2 z = make_float2(0.f, 0.f), o = make_float2(1.f, 0.f);
    applyCtrl(U, tid, w0, w1, z, o, o, z); // CNOT w0->w1
    applyCtrl(U, tid, w1, w2, z, o, o, z); // CNOT w1->w2
    applyCtrl(U, tid, w2, w0, z, o, o, z); // CNOT w2->w0
}

__global__ __launch_bounds__(128) void build_gates(const float* __restrict__ inputs_flat,
                                                   const float* __restrict__ inits,
                                                   const float* __restrict__ s0,
                                                   const float* __restrict__ s1,
                                                   const float* __restrict__ up,
                                                   float* __restrict__ ws) {
    __shared__ float2 U[256];
    const int tid = threadIdx.x;
    float2 a, b, c2, d;

    // ---- per-edge fused 16x16 unitary; local bits: edge=3, neighbor=2, anc1=1, anc2=0
    mat_identity(U, tid);
    rx_gate(inits[0], a, b, c2, d); applyCtrl(U, tid, 2, 1, a, b, c2, d); // ctrl=nb, tgt=anc1
    ry_gate(inits[1], a, b, c2, d); applyCtrl(U, tid, 3, 1, a, b, c2, d); // ctrl=edge, tgt=anc1
    rz_gate(inits[2], a, b, c2, d); applyCtrl(U, tid, 2, 0, a, b, c2, d); // ctrl=nb, tgt=anc2
    ry_gate(inits[3], a, b, c2, d); applyCtrl(U, tid, 3, 0, a, b, c2, d); // ctrl=edge, tgt=anc2
    strong3(U, tid, 3, 2, 1, s0); // [edge, neighbor, anc1]
    strong3(U, tid, 1, 2, 0, s1); // [anc1, neighbor, anc2]
    for (int e = tid; e < 256; e += 128) {
        ws[WS_EDGE_UR + e] = U[e].x;
        ws[WS_EDGE_UI + e] = U[e].y;
    }
    __syncthreads();

    // ---- update 8x8 strong block embedded as U8 (x) I2; local bits: center=3, anc1=2, anc2=1, dummy=0
    mat_identity(U, tid);
    strong3(U, tid, 3, 2, 1, up);
    for (int e = tid; e < 256; e += 128) {
        ws[WS_UPD_UR + e] = U[e].x;
        ws[WS_UPD_UI + e] = U[e].y;
    }

    // ---- per-qubit product-state amplitudes: psi_q = RZ(phi)·RY(theta)|0>
    if (tid < 19) {
        float2* amp = (float2*)(ws + WS_AMP);
        if (tid >= 17) { // ancillas stay |0>
            amp[2 * tid + 0] = make_float2(1.f, 0.f);
            amp[2 * tid + 1] = make_float2(0.f, 0.f);
        } else {
            float th, ph;
            if (tid < 8) { th = inputs_flat[2 * tid]; ph = inputs_flat[2 * tid + 1]; }
            else { th = inputs_flat[16 + 2 * (tid - 8)]; ph = inputs_flat[16 + 2 * (tid - 8) + 1]; }
            float ct = cosf(th * 0.5f), st = sinf(th * 0.5f);
            float cp = cosf(ph * 0.5f), sp = sinf(ph * 0.5f);
            amp[2 * tid + 0] = make_float2(ct * cp, -ct * sp); // cos(t/2) e^{-i p/2}
            amp[2 * tid + 1] = make_float2(st * cp, st * sp);  // sin(t/2) e^{+i p/2}
        }
    }
}

// ================= product-state initialization (one pass, 4 MiB write) =================
__global__ __launch_bounds__(256) void init_state(float* __restrict__ ws) {
    __shared__ float2 samp[38];
    if (threadIdx.x < 38) samp[threadIdx.x] = ((const float2*)(ws + WS_AMP))[threadIdx.x];
    __syncthreads();
    unsigned idx = blockIdx.x * blockDim.x + threadIdx.x; // 2^19 threads
    // wire w <-> flat bit (18 - w)
    float2 acc = samp[(idx >> 18) & 1];
#pragma unroll
    for (int w = 1; w < 19; w++) {
        int bit = (idx >> (18 - w)) & 1;
        acc = cmul(acc, samp[2 * w + bit]);
    }
    ((float2*)(ws + WS_PSI))[idx] = acc;
}

// ================= 4-qubit (16x16 complex) gate via V_WMMA_F32_16X16X4_F32 =================
// Out(16x16) = U x X, 16 state-groups per wave as columns. Complex = 4 real GEMMs,
// each real GEMM = 4 chained K=4 WMMAs -> 16 v_wmma per wave-tile. In-place safe:
// each 16-amplitude group is gathered/scattered by exactly one wave.
__global__ __launch_bounds__(256) void apply_gate16(float* __restrict__ ws, int uofs,
                                                    int b3, int b2, int b1, int b0) {
    const float* Ur = ws + uofs;
    const float* Ui = ws + uofs + 256;
    float2* psi = (float2*)(ws + WS_PSI);

    const int lane = threadIdx.x & 31;
    const unsigned wave = (blockIdx.x * blockDim.x + threadIdx.x) >> 5; // 0..2047
    const int hi = lane >> 4;
    const int n  = lane & 15;

    // insert zeros at the 4 (sorted ascending) target bits to form the group base index
    int s[4] = {b3, b2, b1, b0};
    for (int i = 0; i < 3; i++)
        for (int j = 0; j < 3 - i; j++)
            if (s[j] > s[j + 1]) { int t = s[j]; s[j] = s[j + 1]; s[j + 1] = t; }
    unsigned rest = wave * 16u + (unsigned)n; // group id for this column
#pragma unroll
    for (int i = 0; i < 4; i++) {
        unsigned p = (unsigned)s[i];
        unsigned low = rest & ((1u << p) - 1u);
        rest = ((rest >> p) << (p + 1)) | low;
    }

    auto loff = [&](int l) -> unsigned {
        return ((unsigned)((l >> 3) & 1) << b3) | ((unsigned)((l >> 2) & 1) << b2) |
               ((unsigned)((l >> 1) & 1) << b1) | ((unsigned)(l & 1) << b0);
    };

    const int M = n; // A-matrix row held by this lane (lanes 0-15 and 16-31 both cover M=0..15)
    v8f Dr = {0.f, 0.f, 0.f, 0.f, 0.f, 0.f, 0.f, 0.f};
    v8f Di = {0.f, 0.f, 0.f, 0.f, 0.f, 0.f, 0.f, 0.f};

#pragma unroll
    for (int kk = 0; kk < 4; kk++) {
        const int k0 = 4 * kk + 2 * hi; // VGPR0 K index (lanes 0-15: +0, lanes 16-31: +2)
        const int k1 = k0 + 1;          // VGPR1 K index
        float2 xb0 = psi[rest + loff(k0)];
        float2 xb1 = psi[rest + loff(k1)];
        v2f Br = {xb0.x, xb1.x};
        v2f Bi = {xb0.y, xb1.y};
        v2f Ar = {Ur[M * 16 + k0], Ur[M * 16 + k1]};
        v2f Ai = {Ui[M * 16 + k0], Ui[M * 16 + k1]};
        v2f An = -Ai; // F32 WMMA NEG only supports C-negate; negate Ui in VALU
        // Out_r += Ur*Xr - Ui*Xi ; Out_i += Ur*Xi + Ui*Xr
        Dr = __builtin_amdgcn_wmma_f32_16x16x4_f32(false, Ar, false, Br, (short)0, Dr, false, false);
        Dr = __builtin_amdgcn_wmma_f32_16x16x4_f32(false, An, false, Bi, (short)0, Dr, false, false);
        Di = __builtin_amdgcn_wmma_f32_16x16x4_f32(false, Ar, false, Bi, (short)0, Di, false, false);
        Di = __builtin_amdgcn_wmma_f32_16x16x4_f32(false, Ai, false, Br, (short)0, Di, false, false);
    }

    // D layout: VGPR v -> row M = v + 8*hi, column N = lane&15
#pragma unroll
    for (int v = 0; v < 8; v++) {
        int m = v + 8 * hi;
        psi[rest + loff(m)] = make_float2(Dr[v], Di[v]);
    }
}

// ================= reduction: <Z> on flat bit 8 (H on bit 10 provably drops out) ========
__global__ __launch_bounds__(256) void reduce1(float* __restrict__ ws) {
    const float2* psi = (const float2*)(ws + WS_PSI);
    float acc = 0.f;
    for (unsigned idx = blockIdx.x * blockDim.x + threadIdx.x; idx < (1u << 19);
         idx += gridDim.x * blockDim.x) {
        float2 a = psi[idx];
        float p = fmaf(a.x, a.x, a.y * a.y);
        acc += ((idx >> 8) & 1) ? -p : p;
    }
    __shared__ float red[256];
    red[threadIdx.x] = acc;
    __syncthreads();
    for (int st = 128; st > 0; st >>= 1) {
        if (threadIdx.x < st) red[threadIdx.x] += red[threadIdx.x + st];
        __syncthreads();
    }
    if (threadIdx.x == 0) ws[WS_PART + blockIdx.x] = red[0];
}

__global__ __launch_bounds__(256) void reduce2(const float* __restrict__ ws, float* __restrict__ out) {
    __shared__ float red[256];
    red[threadIdx.x] = ws[WS_PART + threadIdx.x];
    __syncthreads();
    for (int st = 128; st > 0; st >>= 1) {
        if (threadIdx.x < st) red[threadIdx.x] += red[threadIdx.x + st];
        __syncthreads();
    }
    if (threadIdx.x == 0) out[0] = red[0];
}

extern "C" void kernel_launch(void* const* d_in, const int* in_sizes, int n_in,
                              void* d_out, int out_size, void* d_ws, size_t ws_size,
                              hipStream_t stream) {
    const float* inputs_flat = (const float*)d_in[0]; // 34
    const float* inits       = (const float*)d_in[1]; // 4
    const float* s0          = (const float*)d_in[2]; // 9
    const float* s1          = (const float*)d_in[3]; // 9
    const float* up          = (const float*)d_in[4]; // 9
    float* ws  = (float*)d_ws;
    float* out = (float*)d_out;

    build_gates<<<1, 128, 0, stream>>>(inputs_flat, inits, s0, s1, up, ws);
    init_state<<<2048, 256, 0, stream>>>(ws);

    // 8 per-edge 4-qubit gates (same matrix, different wires):
    // wire w <-> flat bit (18-w): edge i -> 18-i, neighbor 9+i -> 9-i, anc1 -> 1, anc2 -> 0
    for (int i = 0; i < 8; i++)
        apply_gate16<<<256, 256, 0, stream>>>(ws, WS_EDGE_UR, 18 - i, 9 - i, 1, 0);

    // update strong block on (center=bit10, anc1=bit1, anc2=bit0), dummy qubit = bit 18
    apply_gate16<<<256, 256, 0, stream>>>(ws, WS_UPD_UR, 10, 1, 0, 18);

    reduce1<<<256, 256, 0, stream>>>(ws);
    reduce2<<<1, 256, 0, stream>>>(ws, out);
}